// _ECELoss_9835475108548
// MI455X (gfx1250) — compile-verified
//
#include <hip/hip_runtime.h>
#include <math.h>

#define ECE_NBINS 15
#define WPB 8            // wave32s per 256-thread block

// ---------------------------------------------------------------------------
// Kernel 0: zero the workspace accumulators (2*NBINS floats).
// ---------------------------------------------------------------------------
__global__ void _ece_zero_bins(float* __restrict__ bins) {
    int t = threadIdx.x;
    if (t < 2 * ECE_NBINS) bins[t] = 0.0f;
}

// ---------------------------------------------------------------------------
// Kernel 1: one wave32 per row, rows staged through LDS with the CDNA5 async
// copy engine (GLOBAL_LOAD_ASYNC_TO_LDS_B128 + S_WAIT_ASYNCCNT), double
// buffered so the next row's HBM fetch overlaps the current row's compute.
// Lanes 0..24 each move one float4 (whole wave = one contiguous 400B row).
// ---------------------------------------------------------------------------
__global__ void __launch_bounds__(256)
_ece_row_kernel(const float* __restrict__ logits,
                const long long* __restrict__ labels,
                float* __restrict__ bins,   // [NBINS conf_sum][NBINS acc_sum]
                int N, int C) {
    __shared__ float  sConf[ECE_NBINS];
    __shared__ float  sAcc[ECE_NBINS];
    __shared__ float4 sBuf[WPB][2][26];     // per-wave double buffer, 16B aligned

    const int tid = threadIdx.x;
    if (tid < ECE_NBINS) { sConf[tid] = 0.0f; sAcc[tid] = 0.0f; }
    __syncthreads();

    const int  lane     = tid & 31;
    const int  wid      = tid >> 5;
    const int  vecCount = C >> 2;           // float4s per row (C=100 -> 25)
    const bool active   = lane < vecCount;

    long long row             = (long long)blockIdx.x * WPB + wid;
    const long long rowStride = (long long)gridDim.x * WPB;

    if (row < N) {
        // ---- prologue: async-stage first row into buffer 0 ----
        if (active) {
            unsigned long long g = (unsigned long long)(uintptr_t)
                ((const float4*)(logits + row * (long long)C) + lane);
            unsigned int l = (unsigned int)(uintptr_t)(void*)&sBuf[wid][0][lane];
            asm volatile("global_load_async_to_lds_b128 %0, %1, off"
                         :: "v"(l), "v"(g) : "memory");
        }

        int p = 0;
        for (; row < N; row += rowStride, p ^= 1) {
            // ---- issue async fetch of the NEXT row into the other buffer ----
            long long nrow = row + rowStride;
            long long frow = (nrow < N) ? nrow : row;   // clamp tail: harmless refetch
            if (active) {
                unsigned long long g = (unsigned long long)(uintptr_t)
                    ((const float4*)(logits + frow * (long long)C) + lane);
                unsigned int l = (unsigned int)(uintptr_t)(void*)&sBuf[wid][p ^ 1][lane];
                asm volatile("global_load_async_to_lds_b128 %0, %1, off"
                             :: "v"(l), "v"(g) : "memory");
                // deeper HBM prefetch (gfx1250 global_prefetch_b8)
                long long pf = row + 2 * rowStride;
                if (pf < N) {
                    __builtin_prefetch((const void*)
                        ((const float4*)(logits + pf * (long long)C) + lane), 0, 0);
                }
            }

            // ---- wait for CURRENT buffer: async loads retire in order, so
            //      <=1 outstanding means buf[p] has landed in LDS ----
            asm volatile("s_wait_asynccnt 0x1" ::: "memory");

            float4 v;
            if (active) {
                v = sBuf[wid][p][lane];     // ds_load_b128
            } else {
                v = make_float4(-INFINITY, -INFINITY, -INFINITY, -INFINITY);
            }

            // Per-lane max + first-index argmax.
            const int base = lane << 2;
            float m  = v.x; int mi = base;
            if (v.y > m) { m = v.y; mi = base + 1; }
            if (v.z > m) { m = v.z; mi = base + 2; }
            if (v.w > m) { m = v.w; mi = base + 3; }

            // Wave32 argmax reduction (first occurrence wins ties, like jnp.argmax).
            #pragma unroll
            for (int off = 16; off > 0; off >>= 1) {
                float om = __shfl_xor(m,  off, 32);
                int   oi = __shfl_xor(mi, off, 32);
                if (om > m || (om == m && oi < mi)) { m = om; mi = oi; }
            }

            // Per-lane sum of exp(x - max), then wave32 sum reduction.
            float s = 0.0f;
            if (active) {
                s = __expf(v.x - m) + __expf(v.y - m) + __expf(v.z - m) + __expf(v.w - m);
            }
            #pragma unroll
            for (int off = 16; off > 0; off >>= 1) {
                s += __shfl_xor(s, off, 32);
            }

            if (lane == 0) {
                const float conf = 1.0f / s;                       // top-1 softmax prob
                int bin = (int)ceilf(conf * (float)ECE_NBINS) - 1; // (b/15,(b+1)/15]
                bin = bin < 0 ? 0 : (bin > ECE_NBINS - 1 ? ECE_NBINS - 1 : bin);
                const float acc = (labels[row] == (long long)mi) ? 1.0f : 0.0f;
                atomicAdd(&sConf[bin], conf);                      // ds_add_f32
                atomicAdd(&sAcc[bin],  acc);
            }
        }
    }

    __syncthreads();
    if (tid < ECE_NBINS) {
        atomicAdd(&bins[tid],             sConf[tid]);             // global_atomic_add_f32
        atomicAdd(&bins[ECE_NBINS + tid], sAcc[tid]);
    }
}

// ---------------------------------------------------------------------------
// Kernel 2: fold 15 bins -> ece. (count==0 bins contribute 0 automatically
// since both sums are 0, so the reference's where(count>0) is redundant.)
// ---------------------------------------------------------------------------
__global__ void _ece_final(const float* __restrict__ bins,
                           float* __restrict__ out, float invN) {
#if __has_builtin(__builtin_amdgcn_s_wait_tensorcnt)
    __builtin_amdgcn_s_wait_tensorcnt(0);   // CDNA5 split-counter path (no-op wait)
#endif
    if (threadIdx.x == 0) {
        float e = 0.0f;
        #pragma unroll
        for (int b = 0; b < ECE_NBINS; ++b) {
            e += fabsf(bins[b] - bins[ECE_NBINS + b]);
        }
        out[0] = e * invN;
    }
}

// ---------------------------------------------------------------------------
// Host-side launch.
// ---------------------------------------------------------------------------
extern "C" void kernel_launch(void* const* d_in, const int* in_sizes, int n_in,
                              void* d_out, int out_size, void* d_ws, size_t ws_size,
                              hipStream_t stream) {
    const float*     logits = (const float*)d_in[0];
    const long long* labels = (const long long*)d_in[1];   // jnp.int64

    const int N = in_sizes[1];                 // 500000
    const int C = (N > 0) ? (in_sizes[0] / N)  // 100
                          : 0;

    float* bins = (float*)d_ws;                // 2*NBINS floats of scratch

    _ece_zero_bins<<<1, 64, 0, stream>>>(bins);

    const int threads = 256;                   // 8 wave32s per block
    long long wantBlocks = ((long long)N + WPB - 1) / WPB;
    int blocks = (int)(wantBlocks < 2048 ? wantBlocks : 2048);
    if (blocks < 1) blocks = 1;

    _ece_row_kernel<<<blocks, threads, 0, stream>>>(logits, labels, bins, N, C);

    _ece_final<<<1, 32, 0, stream>>>(bins, (float*)d_out, 1.0f / (float)N);
}